// STEIConvNetMScaleCompactGRAD_PROP_55662776156164
// MI455X (gfx1250) — compile-verified
//
#include <hip/hip_runtime.h>

#define H 64
#define EDGE_DIM 8
#define DC 16
#define HC 80
#define TT 4

typedef __attribute__((ext_vector_type(2))) float v2f;
typedef __attribute__((ext_vector_type(8))) float v8f;

// ---------------------------------------------------------------------------
// Node projection via V_WMMA_F32_16X16X4_F32, compile-time specialized.
// Y1 = X @ W1 (and Y2 = X @ W2 if DUAL), X:[n_rows,K] W:[K,NC] row-major.
// One wave per 16-row tile. A-fragments for the tile are loaded once into
// registers and reused across all NC/16 column tiles; full unroll lets the
// backend clause the B loads and overlap them with WMMA issue.
// Fragment layouts per CDNA5 ISA 7.12.2:
//   A 16x4 : lane%16 = M, K = vgpr + 2*(lane>=16)      -> v2f contiguous load
//   B 4x16 : lane%16 = N, K = vgpr + 2*(lane>=16)
//   C 16x16: lane%16 = N, M = vgpr + 8*(lane>=16)
// ---------------------------------------------------------------------------
template<int K, int NC, bool DUAL>
__global__ void wmma_proj(const float* __restrict__ X,
                          const float* __restrict__ W1,
                          const float* __restrict__ W2,
                          float* __restrict__ Y1,
                          float* __restrict__ Y2,
                          int n_rows)
{
    int wave = (blockIdx.x * blockDim.x + threadIdx.x) >> 5;
    int lane = threadIdx.x & 31;
    int row_tiles = n_rows >> 4;
    if (wave >= row_tiles) return;          // wave-uniform: EXEC all-1s for WMMA
    int lm = lane & 15;
    int lh = lane >> 4;
    int m0 = wave << 4;

    // A fragments for this row tile, loaded once.
    v2f a[K / 4];
    {
        const float* xrow = X + (size_t)(m0 + lm) * K + 2 * lh;
#pragma unroll
        for (int kk = 0; kk < K / 4; ++kk)
            a[kk] = *(const v2f*)(xrow + 4 * kk);
    }

#pragma unroll
    for (int nt = 0; nt < NC; nt += 16) {
        v8f c1 = {0.f,0.f,0.f,0.f,0.f,0.f,0.f,0.f};
        v8f c2 = {0.f,0.f,0.f,0.f,0.f,0.f,0.f,0.f};
        const float* w1p = W1 + nt + lm + (size_t)2 * lh * NC;
        const float* w2p = W2 + nt + lm + (size_t)2 * lh * NC;
#pragma unroll
        for (int kk = 0; kk < K / 4; ++kk) {
            v2f b1;
            b1.x = w1p[(size_t)(4 * kk) * NC];
            b1.y = w1p[(size_t)(4 * kk + 1) * NC];
            c1 = __builtin_amdgcn_wmma_f32_16x16x4_f32(
                    false, a[kk], false, b1, (short)0, c1, false, false);
            if (DUAL) {
                v2f b2;
                b2.x = w2p[(size_t)(4 * kk) * NC];
                b2.y = w2p[(size_t)(4 * kk + 1) * NC];
                c2 = __builtin_amdgcn_wmma_f32_16x16x4_f32(
                        false, a[kk], false, b2, (short)0, c2, false, false);
            }
        }
#pragma unroll
        for (int r = 0; r < 8; ++r) {
            int rr = m0 + r + 8 * lh;
            Y1[(size_t)rr * NC + nt + lm] = c1[r];
            if (DUAL) Y2[(size_t)rr * NC + nt + lm] = c2[r];
        }
    }
}

// ---------------------------------------------------------------------------
// Edge message + segment-sum for H(=64)-wide layers (IN and OUT layers).
// One wave per edge, each lane handles 2 output channels.
// m = relu(A[src] + B[dst] + ef @ We + rain[src]*Wr); atomic add into acc[dst].
// The 8 edge-feature channels are loaded once by lanes 0..7 and broadcast
// via lane shuffles.
// ---------------------------------------------------------------------------
__global__ void edge_msg_h(const float* __restrict__ A, const float* __restrict__ B,
                           const float* __restrict__ ef,    // e_feats + t; (e*8+j)*TT
                           const float* __restrict__ We,    // [8][64]
                           const float* __restrict__ Wr,    // [64]
                           const float* __restrict__ rain,  // rain0 + t; n*TT
                           const int* __restrict__ src, const int* __restrict__ dst,
                           float* __restrict__ acc, int n_edges)
{
    __shared__ float sWe[EDGE_DIM * H];
    __shared__ float sWr[H];
    for (int i = threadIdx.x; i < EDGE_DIM * H; i += blockDim.x) sWe[i] = We[i];
    if (threadIdx.x < H) sWr[threadIdx.x] = Wr[threadIdx.x];
    __syncthreads();

    int lane = threadIdx.x & 31;
    int gwave = (blockIdx.x * blockDim.x + threadIdx.x) >> 5;
    int nwaves = (gridDim.x * blockDim.x) >> 5;
    int c = lane << 1;

    for (int e = gwave; e < n_edges; e += nwaves) {
        int s = src[e], d = dst[e];
        float rs = rain[(size_t)s * TT];
        // lanes 0..7 fetch the 8 edge-feature channels; broadcast via shuffle
        float efv = (lane < EDGE_DIM) ? ef[((size_t)e * EDGE_DIM + lane) * TT] : 0.f;
        v2f a = *(const v2f*)(A + (size_t)s * H + c);
        v2f b = *(const v2f*)(B + (size_t)d * H + c);
        float m0 = a.x + b.x + rs * sWr[c];
        float m1 = a.y + b.y + rs * sWr[c + 1];
#pragma unroll
        for (int j = 0; j < EDGE_DIM; ++j) {
            float efj = __shfl(efv, j, 32);
            m0 = fmaf(efj, sWe[j * H + c], m0);
            m1 = fmaf(efj, sWe[j * H + c + 1], m1);
        }
        m0 = fmaxf(m0, 0.f);
        m1 = fmaxf(m1, 0.f);
        unsafeAtomicAdd(&acc[(size_t)d * H + c], m0);
        unsafeAtomicAdd(&acc[(size_t)d * H + c + 1], m1);
    }
}

// ---------------------------------------------------------------------------
// Edge message + segment-sum for DC(=16)-wide PROP layers.
// 16 threads per edge (2 edges per wave); edge features either tiled from
// e_feats (node_mode=0) or gathered from previous prop output at src (=1).
// ef vector shared across the 16-thread group via lane shuffles.
// ---------------------------------------------------------------------------
__global__ void edge_msg_p(const float* __restrict__ A, const float* __restrict__ B, // [N,16]
                           const float* __restrict__ ef_t,    // e_feats + t (tiled mode)
                           const float* __restrict__ ef_node, // [N,16] (node mode)
                           const float* __restrict__ We,      // [16][16]
                           const float* __restrict__ Wr,      // [16]
                           const float* __restrict__ rain,
                           const int* __restrict__ src, const int* __restrict__ dst,
                           float* __restrict__ acc, int n_edges, int node_mode)
{
    __shared__ float sWe[DC * DC];
    __shared__ float sWr[DC];
    for (int i = threadIdx.x; i < DC * DC; i += blockDim.x) sWe[i] = We[i];
    if (threadIdx.x < DC) sWr[threadIdx.x] = Wr[threadIdx.x];
    __syncthreads();

    int tid = blockIdx.x * blockDim.x + threadIdx.x;
    int nthr = gridDim.x * blockDim.x;
    int c = tid & (DC - 1);

    for (int idx = tid; idx < n_edges * DC; idx += nthr) {
        int e = idx >> 4;
        int s = src[e], d = dst[e];
        float rs = rain[(size_t)s * TT];
        float myef = node_mode ? ef_node[(size_t)s * DC + c]
                               : ef_t[((size_t)e * EDGE_DIM + (c & 7)) * TT];
        float m = A[(size_t)s * DC + c] + B[(size_t)d * DC + c] + rs * sWr[c];
#pragma unroll
        for (int j = 0; j < DC; ++j) {
            float efj = __shfl(myef, j, 16);   // col-j holder within 16-group
            m = fmaf(efj, sWe[j * DC + c], m);
        }
        m = fmaxf(m, 0.f);
        unsafeAtomicAdd(&acc[(size_t)d * DC + c], m);
    }
}

__global__ void zero_k(float* __restrict__ p, int n) {
    int i = blockIdx.x * blockDim.x + threadIdx.x;
    if (i < n) p[i] = 0.f;
}

__global__ void relu_copy_k(const float* __restrict__ in, float* __restrict__ out, int n) {
    int i = blockIdx.x * blockDim.x + threadIdx.x;
    if (i < n) out[i] = fmaxf(in[i], 0.f);
}

__global__ void concat_k(const float* __restrict__ h, const float* __restrict__ p,
                         float* __restrict__ hout, int n_nodes) {
    int i = blockIdx.x * blockDim.x + threadIdx.x;
    if (i < n_nodes * HC) {
        int n = i / HC, c = i - n * HC;
        hout[i] = (c < H) ? h[(size_t)n * H + c] : p[(size_t)n * DC + (c - H)];
    }
}

// rain_out[n] = dot(h[n,:], W_rain); one wave per node, butterfly reduction.
__global__ void rain_k(const float* __restrict__ h, const float* __restrict__ Wr,
                       float* __restrict__ out, int n_nodes, int t)
{
    int lane = threadIdx.x & 31;
    int gwave = (blockIdx.x * blockDim.x + threadIdx.x) >> 5;
    int nwaves = (gridDim.x * blockDim.x) >> 5;
    for (int n = gwave; n < n_nodes; n += nwaves) {
        int c = lane << 1;
        float p = h[(size_t)n * H + c] * Wr[c] + h[(size_t)n * H + c + 1] * Wr[c + 1];
        for (int off = 16; off > 0; off >>= 1) p += __shfl_down(p, off, 32);
        if (lane == 0) out[(size_t)n * TT + t] = p;
    }
}

extern "C" void kernel_launch(void* const* d_in, const int* in_sizes, int n_in,
                              void* d_out, int out_size, void* d_ws, size_t ws_size,
                              hipStream_t stream)
{
    const float* inputs  = (const float*)d_in[0];
    const float* e_feats = (const float*)d_in[1];
    const float* rain0   = (const float*)d_in[2];
    const int*   src     = (const int*)d_in[3];
    const int*   dst     = (const int*)d_in[4];
    const float* W_emb   = (const float*)d_in[5];
    const float* Win_hs  = (const float*)d_in[6];
    const float* Win_hd  = (const float*)d_in[7];
    const float* Win_e   = (const float*)d_in[8];
    const float* Win_r   = (const float*)d_in[9];
    const float* Wp_hs   = (const float*)d_in[10];
    const float* Wp_hd   = (const float*)d_in[11];
    const float* Wp_e    = (const float*)d_in[12];
    const float* Wp_r    = (const float*)d_in[13];
    const float* Wo_hs   = (const float*)d_in[14];
    const float* Wo_hd   = (const float*)d_in[15];
    const float* Wo_e    = (const float*)d_in[16];
    const float* Wo_r    = (const float*)d_in[17];
    const float* W_rain  = (const float*)d_in[18];
    float* out = (float*)d_out;

    const int N_ = in_sizes[2] / TT;   // 40000
    const int E_ = in_sizes[3];        // 320000

    float* ws = (float*)d_ws;
    float* h    = ws; ws += (size_t)N_ * H;
    float* Aq   = ws; ws += (size_t)N_ * H;
    float* Bq   = ws; ws += (size_t)N_ * H;
    float* acc  = ws; ws += (size_t)N_ * H;
    float* hout = ws; ws += (size_t)N_ * HC;
    float* Ap0  = ws; ws += (size_t)N_ * DC;
    float* Bp0  = ws; ws += (size_t)N_ * DC;
    float* Ap1  = ws; ws += (size_t)N_ * DC;
    float* Bp1  = ws; ws += (size_t)N_ * DC;
    float* p0   = ws; ws += (size_t)N_ * DC;
    float* p1   = ws; ws += (size_t)N_ * DC;

    const dim3 blk(256);
    const int mmBlocks  = ((N_ / 16) + 7) / 8;          // 8 waves/block, 1 tile/wave
    const int ehBlocks  = (E_ + 7) / 8;                 // 1 edge/wave, 8 waves/block
    const int epBlocks  = (E_ * DC + 255) / 256;        // 16 threads/edge
    const int zHBlocks  = (N_ * H + 255) / 256;
    const int zPBlocks  = (N_ * DC + 255) / 256;
    const int ccBlocks  = (N_ * HC + 255) / 256;
    const int rnBlocks  = (N_ + 7) / 8;

    // h = inputs @ W_emb
    wmma_proj<16, H, false><<<mmBlocks, blk, 0, stream>>>(inputs, W_emb, W_emb, h, h, N_);

    for (int t = 0; t < TT; ++t) {
        const float* ef_t   = e_feats + t;
        const float* rain_t = rain0 + t;

        // ---- IN layers (S1 = 2) ----
        for (int s = 0; s < 2; ++s) {
            wmma_proj<H, H, true><<<mmBlocks, blk, 0, stream>>>(
                h, Win_hs + s * H * H, Win_hd + s * H * H, Aq, Bq, N_);
            zero_k<<<zHBlocks, blk, 0, stream>>>(acc, N_ * H);
            edge_msg_h<<<ehBlocks, blk, 0, stream>>>(Aq, Bq, ef_t,
                                                     Win_e + s * EDGE_DIM * H,
                                                     Win_r + s * H,
                                                     rain_t, src, dst, acc, E_);
            relu_copy_k<<<zHBlocks, blk, 0, stream>>>(acc, h, N_ * H);
        }

        // ---- PROP chain ----
        wmma_proj<H, DC, true><<<mmBlocks, blk, 0, stream>>>(
            h, Wp_hs, Wp_hd, Ap0, Bp0, N_);
        wmma_proj<H, DC, true><<<mmBlocks, blk, 0, stream>>>(
            h, Wp_hs + H * DC, Wp_hd + H * DC, Ap1, Bp1, N_);
        int seq_s[3], seq_node[3], nseq;
        if (t == 0) { seq_s[0]=0; seq_node[0]=0; seq_s[1]=0; seq_node[1]=1;
                      seq_s[2]=1; seq_node[2]=1; nseq = 3; }   // faithful quirk
        else        { seq_s[0]=0; seq_node[0]=0; seq_s[1]=1; seq_node[1]=1; nseq = 2; }

        float* pc = p0;
        float* pn = p1;
        for (int i = 0; i < nseq; ++i) {
            int s = seq_s[i];
            zero_k<<<zPBlocks, blk, 0, stream>>>(pn, N_ * DC);
            edge_msg_p<<<epBlocks, blk, 0, stream>>>(s ? Ap1 : Ap0, s ? Bp1 : Bp0,
                                                     ef_t, pc,
                                                     Wp_e + s * DC * DC,
                                                     Wp_r + s * DC,
                                                     rain_t, src, dst, pn, E_,
                                                     seq_node[i]);
            float* tmp = pc; pc = pn; pn = tmp;
        }

        // ---- OUT layer ----
        concat_k<<<ccBlocks, blk, 0, stream>>>(h, pc, hout, N_);
        wmma_proj<HC, H, true><<<mmBlocks, blk, 0, stream>>>(
            hout, Wo_hs, Wo_hd, Aq, Bq, N_);
        zero_k<<<zHBlocks, blk, 0, stream>>>(acc, N_ * H);
        edge_msg_h<<<ehBlocks, blk, 0, stream>>>(Aq, Bq, ef_t, Wo_e, Wo_r,
                                                 rain_t, src, dst, acc, E_);
        relu_copy_k<<<zHBlocks, blk, 0, stream>>>(acc, h, N_ * H);
        rain_k<<<rnBlocks, blk, 0, stream>>>(h, W_rain, out, N_, t);
    }
}